// EntityEncoder_65111704207698
// MI455X (gfx1250) — compile-verified
//
#include <hip/hip_runtime.h>
#include <hip/hip_bf16.h>

typedef float v2f __attribute__((ext_vector_type(2)));
typedef float v8f __attribute__((ext_vector_type(8)));

#define BDIM 8
#define SDIM 4096
#define DDIM 256
#define SEG  128
#define EDIM 32           // SDIM / SEG
#define XSTR 260          // padded LDS stride for 256-wide rows (bank-conflict free frags)
#define SSTR 132          // padded LDS stride for 128-wide score rows
#define MASKVAL (-3.402823466e38f)

__device__ __forceinline__ v8f wmma_f32(v2f a, v2f b, v8f c) {
  // D = A(16x4,f32) * B(4x16,f32) + C(16x16,f32)
  return __builtin_amdgcn_wmma_f32_16x16x4_f32(false, a, false, b, (short)0, c, false, false);
}

// ---- fix[e] = any_{b,l} all_k (mask[b, e*SEG+l, e*SEG+k] == MASKVAL) ----
__global__ void fix_init_kernel(int* fix) {
  if (threadIdx.x < EDIM) fix[threadIdx.x] = 0;
}

__global__ __launch_bounds__(256) void fix_kernel(const float* __restrict__ hmask,
                                                  int* __restrict__ fix) {
  const int gw   = (int)((blockIdx.x * blockDim.x + threadIdx.x) >> 5); // one wave per row
  const int lane = threadIdx.x & 31;
  if (gw >= BDIM * EDIM * SEG) return;
  const int l  = gw & (SEG - 1);
  const int e  = (gw >> 7) & (EDIM - 1);
  const int bb = gw >> 12;
  const float4 v = *(const float4*)(hmask +
      ((size_t)bb * SDIM + (size_t)e * SEG + l) * SDIM + (size_t)e * SEG + lane * 4);
  const bool allm = (v.x == MASKVAL) && (v.y == MASKVAL) && (v.z == MASKVAL) && (v.w == MASKVAL);
  if (__all(allm)) {
    if (lane == 0) atomicOr(&fix[e], 1);
  }
}

// ---- new_mask[b,e,s] = (s/SEG == e) ----
__global__ __launch_bounds__(256) void newmask_kernel(float* __restrict__ out) {
  const size_t idx = (size_t)blockIdx.x * blockDim.x + threadIdx.x;
  if (idx < (size_t)BDIM * EDIM * SDIM) {
    const int s = (int)(idx & (SDIM - 1));
    const int e = (int)((idx >> 12) & (EDIM - 1));
    out[idx] = ((s >> 7) == e) ? 1.0f : 0.0f;
  }
}

// ---- main fused tile kernel: one workgroup per (b,e) tile, 8 wave32s ----
__global__ __launch_bounds__(256) void attn_kernel(
    const float* __restrict__ hidden, const float* __restrict__ hmask,
    const float* __restrict__ w, const float* __restrict__ bias,
    const int* __restrict__ fixE, float* __restrict__ pooled) {
  extern __shared__ float smem[];
  float* Xs  = smem;             // 128 x XSTR (X tile)
  float* Ts  = smem + 128 * XSTR;// 128 x XSTR (tanh activations)
  float* Ssc = Ts;               // 128 x SSTR (scores; aliases Ts after barrier)

  const int be   = blockIdx.x;   // 0..255
  const int bb   = be >> 5;
  const int ee   = be & (EDIM - 1);
  const int tid  = threadIdx.x;
  const int wave = tid >> 5;
  const int lane = tid & 31;
  const int n    = lane & 15;           // N / M-row index inside a 16-wide frag
  const int kp   = (lane >> 4) << 1;    // K sub-offset: 0 for lanes 0-15, 2 for 16-31
  const int mh   = (lane >> 4) << 3;    // C/D row-half offset: 0 or 8

  const float* Xg = hidden + ((size_t)bb * SDIM + (size_t)ee * SEG) * DDIM;

  // Stage X (128x256) into LDS with padded stride.
  for (int idx = tid * 4; idx < SEG * DDIM; idx += 256 * 4) {
    const int r = idx >> 8;
    const int c = idx & (DDIM - 1);
    *(float4*)(Xs + r * XSTR + c) = *(const float4*)(Xg + idx);
  }
  __syncthreads();

  // ---- Phase 1: T = tanh(X * W^T + b). Wave w owns col tiles j = 2w, 2w+1. ----
  for (int jj = 0; jj < 2; ++jj) {
    const int j    = wave * 2 + jj;
    const int mcol = j * 16 + n;
    const float* wrow = w + (size_t)mcol * DDIM;
    v8f zero = {};
    v8f acc[8];
#pragma unroll
    for (int i = 0; i < 8; ++i) acc[i] = zero;
    for (int k = 0; k < DDIM; k += 4) {
      v2f bf;                                  // B[K=d][N=m] = W[m][d]
      bf.x = wrow[k + kp];
      bf.y = wrow[k + kp + 1];
      v2f af[8];                               // batch A-frag loads -> 8 ds_loads in flight
#pragma unroll
      for (int i = 0; i < 8; ++i) {
        const float* xr = Xs + (i * 16 + n) * XSTR + k + kp;
        af[i].x = xr[0]; af[i].y = xr[1];      // A[l][d] = X
      }
#pragma unroll
      for (int i = 0; i < 8; ++i) acc[i] = wmma_f32(af[i], bf, acc[i]);
    }
    const float bv = bias[mcol];
#pragma unroll
    for (int i = 0; i < 8; ++i)
#pragma unroll
      for (int r = 0; r < 8; ++r)
        Ts[(i * 16 + mh + r) * XSTR + mcol] = tanhf(acc[i][r] + bv);
  }
  __syncthreads();

  // ---- Phase 2: S = T * X^T. Wave w owns col tile j = w (k columns 16w..16w+15). ----
  {
    const int j    = wave;
    const int kcol = j * 16 + n;
    v8f zero = {};
    v8f acc[8];
#pragma unroll
    for (int i = 0; i < 8; ++i) acc[i] = zero;
    for (int k = 0; k < DDIM; k += 4) {
      const float* xr = Xs + kcol * XSTR + k + kp; // B[K=m][N=kcol] = X[kcol][m]
      v2f bf; bf.x = xr[0]; bf.y = xr[1];
      v2f af[8];
#pragma unroll
      for (int i = 0; i < 8; ++i) {
        const float* tr = Ts + (i * 16 + n) * XSTR + k + kp;
        af[i].x = tr[0]; af[i].y = tr[1];
      }
#pragma unroll
      for (int i = 0; i < 8; ++i) acc[i] = wmma_f32(af[i], bf, acc[i]);
    }
    __syncthreads();  // everyone finished reading Ts; safe to overwrite with scores
#pragma unroll
    for (int i = 0; i < 8; ++i)
#pragma unroll
      for (int r = 0; r < 8; ++r)
        Ssc[(i * 16 + mh + r) * SSTR + j * 16 + n] = acc[i][r];
  }
  __syncthreads();

  // ---- Phase 3: P = softmax(S + diag_mask_block) row-wise (with fix rule). ----
  if (tid < SEG) {
    const int l = tid;
    const float* mrow = hmask +
        ((size_t)bb * SDIM + (size_t)ee * SEG + l) * SDIM + (size_t)ee * SEG;
    const bool zrow = (l == 0) && (fixE[ee] != 0);
    float* srow = Ssc + l * SSTR;
    float mx = -3.402823466e38f;
    for (int k0 = 0; k0 < SEG; ++k0) {
      const int k = (k0 + l) & (SEG - 1);          // rotate to stagger LDS banks
      const float m = zrow ? 0.0f : mrow[k];
      mx = fmaxf(mx, srow[k] + m);
    }
    float sum = 0.0f;
    for (int k0 = 0; k0 < SEG; ++k0) {
      const int k = (k0 + l) & (SEG - 1);
      const float m = zrow ? 0.0f : mrow[k];
      const float v = expf(srow[k] + m - mx);
      srow[k] = v;
      sum += v;
    }
    const float inv = 1.0f / sum;
    for (int k0 = 0; k0 < SEG; ++k0) {
      const int k = (k0 + l) & (SEG - 1);
      srow[k] *= inv;
    }
  }
  __syncthreads();

  // ---- Phase 4: ctx = P * X, then pooled[d] = max_l ctx[l][d] (in registers). ----
  for (int jj = 0; jj < 2; ++jj) {
    const int j    = wave * 2 + jj;
    const int dcol = j * 16 + n;
    v8f zero = {};
    v8f acc[8];
#pragma unroll
    for (int i = 0; i < 8; ++i) acc[i] = zero;
    for (int k = 0; k < SEG; k += 4) {
      v2f bf;                                    // B[K=k][N=d] = X[k][d]
      bf.x = Xs[(k + kp) * XSTR + dcol];
      bf.y = Xs[(k + kp + 1) * XSTR + dcol];
      v2f af[8];
#pragma unroll
      for (int i = 0; i < 8; ++i) {
        const float* pr = Ssc + (i * 16 + n) * SSTR + k + kp;
        af[i].x = pr[0]; af[i].y = pr[1];        // A[l][k] = P
      }
#pragma unroll
      for (int i = 0; i < 8; ++i) acc[i] = wmma_f32(af[i], bf, acc[i]);
    }
    // reduce max over all 128 rows: over 8 VGPRs, 8 row tiles, then lane^16 halves
    v8f cm = acc[0];
#pragma unroll
    for (int i = 1; i < 8; ++i)
#pragma unroll
      for (int r = 0; r < 8; ++r) cm[r] = fmaxf(cm[r], acc[i][r]);
    float m = cm[0];
#pragma unroll
    for (int r = 1; r < 8; ++r) m = fmaxf(m, cm[r]);
    m = fmaxf(m, __shfl_xor(m, 16, 32));
    if (lane < 16) pooled[(size_t)be * DDIM + dcol] = m;
  }
}

extern "C" void kernel_launch(void* const* d_in, const int* in_sizes, int n_in,
                              void* d_out, int out_size, void* d_ws, size_t ws_size,
                              hipStream_t stream) {
  (void)in_sizes; (void)n_in; (void)out_size; (void)ws_size;
  const float* hidden = (const float*)d_in[0];
  const float* hmask  = (const float*)d_in[1];
  const float* w      = (const float*)d_in[2];
  const float* bias   = (const float*)d_in[3];
  // d_in[4] = seg_len (128, hardcoded)

  float* pooled = (float*)d_out;                          // (8,32,256)
  float* nmask  = (float*)d_out + BDIM * EDIM * DDIM;     // (8,32,4096)
  int*   fix    = (int*)d_ws;                             // 32 ints

  fix_init_kernel<<<1, 32, 0, stream>>>(fix);
  fix_kernel<<<(BDIM * EDIM * SEG * 32 + 255) / 256, 256, 0, stream>>>(hmask, fix);

  const size_t lds_bytes = (size_t)(2 * 128 * XSTR) * sizeof(float); // 266,240 B
  attn_kernel<<<BDIM * EDIM, 256, lds_bytes, stream>>>(hidden, hmask, w, bias, fix, pooled);

  newmask_kernel<<<(BDIM * EDIM * SDIM + 255) / 256, 256, 0, stream>>>(nmask);
}